// MpnnModel_57140244906225
// MI455X (gfx1250) — compile-verified
//
#include <hip/hip_runtime.h>
#include <math.h>

// ---------------------------------------------------------------------------
// MPNN (Gilmer-style) on MI455X / gfx1250, fp32 WMMA (V_WMMA_F32_16X16X4_F32).
//
// Key algebraic rewrite: the per-edge message GEMM + masked sum collapses to
//   agg[b,i] = [deg_i * h_i | (adj @ h)_i | Eagg_i] @ W_msg[l] + deg_i * b_msg[l]
// with adj/deg/Eagg precomputed once from `edges` (the only 32MB read).
// All GEMMs run on wave-per-16x16-tile fp32 WMMA (exact precision match).
// ---------------------------------------------------------------------------

#define B_ 32
#define N_ 128
#define F_ 64
#define E_ 16
#define M_ 128
#define R_ 256
#define L_ 3
#define H_ 10
#define KIN 144   /* 2F+E */
#define G3 192    /* 3F   */

typedef __attribute__((ext_vector_type(2))) float v2f;
typedef __attribute__((ext_vector_type(8))) float v8f;

__device__ __forceinline__ v8f wmma4(v2f a, v2f b, v8f c) {
  // 8 args: (neg_a, A, neg_b, B, c_mod, C, reuse_a, reuse_b)
  return __builtin_amdgcn_wmma_f32_16x16x4_f32(
      false, a, false, b, (short)0, c, false, false);
}

// C(16x16) tile at (row0, col0):  acc += A[row0+m, :K] * B[:K, col0+n]
// A row-major lda, B row-major ldb.  Fragment layout per ISA 7.12.2:
//   lane L, vgpr v:  A elem [L%16][2*(L/16)+v],  B elem [2*(L/16)+v][L%16]
__device__ __forceinline__ v8f mma_tile(const float* __restrict__ A, int lda, int row0,
                                        const float* __restrict__ Bm, int ldb, int col0,
                                        int K, v8f acc) {
  const int lane = threadIdx.x & 31;
  const int hi = lane >> 4;
  const int mn = lane & 15;
  const float* ap = A + (row0 + mn) * lda;
  const float* bp = Bm + col0 + mn;
  for (int k0 = 0; k0 < K; k0 += 4) {
    int ka = k0 + 2 * hi;
    v2f a; a.x = ap[ka];          a.y = ap[ka + 1];
    v2f b; b.x = bp[ka * ldb];    b.y = bp[(ka + 1) * ldb];
    acc = wmma4(a, b, acc);
  }
  return acc;
}

__device__ __forceinline__ float sigf(float x) { return 1.f / (1.f + __expf(-x)); }

// ---------------- K0: adj mask, degree, masked-edge aggregate (reads 32MB once)
__global__ void k_pre(const float* __restrict__ edges, float* __restrict__ adj,
                      float* __restrict__ deg, float* __restrict__ eagg) {
  __shared__ float s[17 * N_];
  const int row = blockIdx.x;                 // b*N + i
  const int j = threadIdx.x;                  // 0..127
  const float* ep = edges + ((size_t)row * N_ + j) * E_;
  float e[E_];
#pragma unroll
  for (int c = 0; c < E_; ++c) e[c] = ep[c];
  float adjv = (e[0] != 0.f || e[1] != 0.f || e[2] != 0.f || e[3] != 0.f) ? 1.f : 0.f;
  adj[(size_t)row * N_ + j] = adjv;
#pragma unroll
  for (int c = 0; c < E_; ++c) s[c * N_ + j] = adjv * e[c];
  s[16 * N_ + j] = adjv;
  __syncthreads();
  if (j < 17) {
    float sum = 0.f;
    for (int t = 0; t < N_; ++t) sum += s[j * N_ + t];
    if (j < 16) eagg[row * E_ + j] = sum;
    else        deg[row] = sum;
  }
}

// ---------------- K1: Hagg[b] = adj[b] @ h[b]   (batched 128x128x64, WMMA)
__global__ void k_hagg(const float* __restrict__ adj, const float* __restrict__ h,
                       float* __restrict__ hagg) {
  const int id = blockIdx.x;        // b*32 + mt*4 + nt
  const int b = id >> 5, mt = (id >> 2) & 7, nt = id & 3;
  const float* A  = adj + (size_t)b * N_ * N_;
  const float* Bm = h   + (size_t)b * N_ * F_;
  v8f acc;
#pragma unroll
  for (int r = 0; r < 8; ++r) acc[r] = 0.f;
  acc = mma_tile(A, N_, mt * 16, Bm, F_, nt * 16, N_, acc);
  const int lane = threadIdx.x & 31, hi = lane >> 4, mn = lane & 15;
  float* C = hagg + (size_t)b * N_ * F_;
#pragma unroll
  for (int r = 0; r < 8; ++r)
    C[(mt * 16 + r + 8 * hi) * F_ + nt * 16 + mn] = acc[r];
}

// ---------------- K2: X = [deg*h | Hagg | Eagg]   (elementwise)
__global__ void k_buildx(const float* __restrict__ h, const float* __restrict__ hagg,
                         const float* __restrict__ eagg, const float* __restrict__ deg,
                         float* __restrict__ X) {
  int idx = blockIdx.x * blockDim.x + threadIdx.x;
  if (idx >= B_ * N_ * KIN) return;
  int row = idx / KIN, c = idx % KIN;
  float v;
  if (c < F_)            v = deg[row] * h[row * F_ + c];
  else if (c < 2 * F_)   v = hagg[row * F_ + (c - F_)];
  else                   v = eagg[row * E_ + (c - 2 * F_)];
  X[idx] = v;
}

// ---------------- K3: agg = X @ W_msg[l] + deg*b_msg[l]   (4096x144x128, WMMA)
__global__ void k_msg(const float* __restrict__ X, const float* __restrict__ Wm,
                      const float* __restrict__ bm, const float* __restrict__ deg,
                      float* __restrict__ agg) {
  const int id = blockIdx.x;        // mt*8 + nt
  const int mt = id >> 3, nt = id & 7;
  v8f acc;
#pragma unroll
  for (int r = 0; r < 8; ++r) acc[r] = 0.f;
  acc = mma_tile(X, KIN, mt * 16, Wm, M_, nt * 16, KIN, acc);
  const int lane = threadIdx.x & 31, hi = lane >> 4, mn = lane & 15;
  const float bv = bm[nt * 16 + mn];
#pragma unroll
  for (int r = 0; r < 8; ++r) {
    int row = mt * 16 + r + 8 * hi;
    agg[row * M_ + nt * 16 + mn] = acc[r] + deg[row] * bv;
  }
}

// ---------------- K4: fused GRU step (Keras reset_after=True), WMMA for gx/gh
__global__ void k_gru(const float* __restrict__ agg, const float* __restrict__ h,
                      const float* __restrict__ Wx, const float* __restrict__ Wh,
                      const float* __restrict__ bin, const float* __restrict__ brec,
                      float* __restrict__ hout) {
  const int row0 = blockIdx.x * 16;            // 256 row tiles
  const int lane = threadIdx.x & 31, hi = lane >> 4, mn = lane & 15;
#pragma unroll 1
  for (int t = 0; t < 4; ++t) {                // 4 column tiles of F
    const int cz = t * 16, cr = 64 + t * 16, ch = 128 + t * 16;
    v8f az, ar, axh, ahh;
    const float bz  = bin[cz + mn] + brec[cz + mn];
    const float br  = bin[cr + mn] + brec[cr + mn];
    const float bxh = bin[ch + mn];
    const float bhh = brec[ch + mn];
#pragma unroll
    for (int r = 0; r < 8; ++r) { az[r] = bz; ar[r] = br; axh[r] = bxh; ahh[r] = bhh; }
    az  = mma_tile(agg, M_, row0, Wx, G3, cz, M_, az);
    az  = mma_tile(h,   F_, row0, Wh, G3, cz, F_, az);
    ar  = mma_tile(agg, M_, row0, Wx, G3, cr, M_, ar);
    ar  = mma_tile(h,   F_, row0, Wh, G3, cr, F_, ar);
    axh = mma_tile(agg, M_, row0, Wx, G3, ch, M_, axh);
    ahh = mma_tile(h,   F_, row0, Wh, G3, ch, F_, ahh);
#pragma unroll
    for (int r = 0; r < 8; ++r) {
      int row = row0 + r + 8 * hi;
      float hold = h[row * F_ + t * 16 + mn];
      float z  = sigf(az[r]);
      float rr = sigf(ar[r]);
      float hc = tanhf(axh[r] + rr * ahh[r]);
      hout[row * F_ + t * 16 + mn] = z * hold + (1.f - z) * hc;
    }
  }
}

// ---------------- K5: readout[b] = sum_v sigmoid([h,x] Wi + bi) * (h Wj + bj)
__global__ void k_readout(const float* __restrict__ h, const float* __restrict__ nodes,
                          const float* __restrict__ Wi, const float* __restrict__ bi,
                          const float* __restrict__ Wj, const float* __restrict__ bj,
                          float* __restrict__ rd) {
  const int id = blockIdx.x;        // b*16 + nt
  const int b = id >> 4, nt = id & 15;
  const int lane = threadIdx.x & 31, hi = lane >> 4, mn = lane & 15;
  const int col0 = nt * 16;
  const float* hb = h     + (size_t)b * N_ * F_;
  const float* xb = nodes + (size_t)b * N_ * F_;
  float total = 0.f;
  for (int mtile = 0; mtile < 8; ++mtile) {
    const int row0 = mtile * 16;
    v8f ai, aj;
    const float biv = bi[col0 + mn], bjv = bj[col0 + mn];
#pragma unroll
    for (int r = 0; r < 8; ++r) { ai[r] = biv; aj[r] = bjv; }
    // concat([h, x]) @ Wi  ==  h @ Wi[0:64] + x @ Wi[64:128]
    ai = mma_tile(hb, F_, row0, Wi,           R_, col0, F_, ai);
    ai = mma_tile(xb, F_, row0, Wi + F_ * R_, R_, col0, F_, ai);
    aj = mma_tile(hb, F_, row0, Wj,           R_, col0, F_, aj);
#pragma unroll
    for (int r = 0; r < 8; ++r) total += sigf(ai[r]) * aj[r];
  }
  total += __shfl_xor(total, 16, 32);   // combine row-halves (same column)
  if (hi == 0) rd[b * R_ + col0 + mn] = total;
}

// ---------------- K6: logits = readout @ Wc + bc ; softmax
__global__ void k_head(const float* __restrict__ rd, const float* __restrict__ Wc,
                       const float* __restrict__ bc, float* __restrict__ out) {
  const int b = blockIdx.x;
  const int lane = threadIdx.x;
  float part[H_];
#pragma unroll
  for (int c = 0; c < H_; ++c) part[c] = 0.f;
  for (int r = lane; r < R_; r += 32) {
    float rv = rd[b * R_ + r];
#pragma unroll
    for (int c = 0; c < H_; ++c) part[c] += rv * Wc[r * H_ + c];
  }
#pragma unroll
  for (int off = 16; off >= 1; off >>= 1)
#pragma unroll
    for (int c = 0; c < H_; ++c) part[c] += __shfl_xor(part[c], off, 32);
  if (lane == 0) {
    float lg[H_], mx = -1e30f;
#pragma unroll
    for (int c = 0; c < H_; ++c) { lg[c] = part[c] + bc[c]; mx = fmaxf(mx, lg[c]); }
    float s = 0.f;
#pragma unroll
    for (int c = 0; c < H_; ++c) { lg[c] = __expf(lg[c] - mx); s += lg[c]; }
#pragma unroll
    for (int c = 0; c < H_; ++c) out[b * H_ + c] = lg[c] / s;
  }
}

// ---------------------------------------------------------------------------
extern "C" void kernel_launch(void* const* d_in, const int* in_sizes, int n_in,
                              void* d_out, int out_size, void* d_ws, size_t ws_size,
                              hipStream_t stream) {
  const float* nodes = (const float*)d_in[0];
  const float* edges = (const float*)d_in[1];
  const float* W_msg = (const float*)d_in[2];
  const float* b_msg = (const float*)d_in[3];
  const float* Wx    = (const float*)d_in[4];
  const float* Wh    = (const float*)d_in[5];
  const float* b_in  = (const float*)d_in[6];
  const float* b_rec = (const float*)d_in[7];
  const float* Wi    = (const float*)d_in[8];
  const float* bi    = (const float*)d_in[9];
  const float* Wj    = (const float*)d_in[10];
  const float* bj    = (const float*)d_in[11];
  const float* Wc    = (const float*)d_in[12];
  const float* bc    = (const float*)d_in[13];
  float* out = (float*)d_out;

  float* ws   = (float*)d_ws;
  float* adj  = ws; ws += (size_t)B_ * N_ * N_;   // 2 MB
  float* deg  = ws; ws += B_ * N_;
  float* eagg = ws; ws += B_ * N_ * E_;
  float* hagg = ws; ws += B_ * N_ * F_;
  float* X    = ws; ws += B_ * N_ * KIN;
  float* agg  = ws; ws += B_ * N_ * M_;
  float* h0   = ws; ws += B_ * N_ * F_;
  float* h1   = ws; ws += B_ * N_ * F_;
  float* rd   = ws; ws += B_ * R_;

  k_pre<<<B_ * N_, N_, 0, stream>>>(edges, adj, deg, eagg);

  const float* hcur = nodes;
  float* hbuf[2] = {h0, h1};
  for (int l = 0; l < L_; ++l) {
    float* hnext = hbuf[l & 1];
    k_hagg<<<B_ * 32, 32, 0, stream>>>(adj, hcur, hagg);
    const int nX = B_ * N_ * KIN;
    k_buildx<<<(nX + 255) / 256, 256, 0, stream>>>(hcur, hagg, eagg, deg, X);
    k_msg<<<256 * 8, 32, 0, stream>>>(X, W_msg + (size_t)l * KIN * M_,
                                      b_msg + l * M_, deg, agg);
    k_gru<<<256, 32, 0, stream>>>(agg, hcur, Wx + (size_t)l * M_ * G3,
                                  Wh + (size_t)l * F_ * G3,
                                  b_in + l * G3, b_rec + l * G3, hnext);
    hcur = hnext;
  }
  k_readout<<<B_ * 16, 32, 0, stream>>>(hcur, nodes, Wi, bi, Wj, bj, rd);
  k_head<<<B_, 32, 0, stream>>>(rd, Wc, bc, out);
}